// DGNNLayer_1211180777852
// MI455X (gfx1250) — compile-verified
//
#include <hip/hip_runtime.h>

#define D_DIM 128   // embedding dim; 1/sqrt(D + HIDDEN_DIM) = 1/sqrt(256) = 1/16

typedef __attribute__((ext_vector_type(2))) float v2f;
typedef __attribute__((ext_vector_type(8))) float v8f;

// Order-preserving float <-> uint mapping so we can use integer atomicMax for
// the segment max (monotone: a < b  <=>  f32_ord(a) < f32_ord(b)).
__device__ __forceinline__ unsigned int f32_ord(float f) {
  unsigned int b = __float_as_uint(f);
  return (b & 0x80000000u) ? ~b : (b | 0x80000000u);
}
__device__ __forceinline__ float f32_unord(unsigned int u) {
  unsigned int b = (u & 0x80000000u) ? (u & 0x7FFFFFFFu) : ~u;
  return __uint_as_float(b);
}

// ---------------------------------------------------------------------------
// K0: zero the output and per-node accumulators (harness poisons buffers).
// ---------------------------------------------------------------------------
__global__ void init_ws_kernel(float* __restrict__ out,
                               float* __restrict__ seg_sum,
                               unsigned int* __restrict__ seg_max_ord,
                               int N) {
  long long i = (long long)blockIdx.x * blockDim.x + threadIdx.x;
  long long total = (long long)N * D_DIM;
  if (i < total) out[i] = 0.0f;
  if (i < N) {
    seg_sum[i] = 0.0f;
    seg_max_ord[i] = 0u;  // below f32_ord of every finite float
  }
}

// ---------------------------------------------------------------------------
// K1: alpha[e] = dot(ent[dst[e]], ent[src[e]]) / 16 via V_WMMA_F32_16X16X4_F32.
// One wave handles 16 edges: C(16x16) += A(16x4) x B(4x16), accumulated over
// 32 K-chunks of 4; diag(C) holds the 16 dot products.
//
// A (16x4 f32) lane layout: lanes 0-15 hold K=0,1 (v[0],v[1]); lanes 16-31
// hold K=2,3. Row M = lane&15. B (4x16) mirrors with N = lane&15. So lane L
// loads float2 at row-offset 4*kk + 2*(L>>4) from the edge-(L&15) vectors.
// D (16x16 f32): lanes 0-15 -> N=lane, M=vgpr r; lanes 16-31 -> N=lane-16,
// M=r+8. Diagonal e: (lane=e, c[e]) for e<8; (lane=e+16, c[e-8]) for e>=8.
// ---------------------------------------------------------------------------
__global__ void __launch_bounds__(256)
alpha_wmma_kernel(const float* __restrict__ ent,
                  const int* __restrict__ src,
                  const int* __restrict__ dst,
                  float* __restrict__ alpha,
                  unsigned int* __restrict__ seg_max_ord,
                  int E) {
  const int lane = threadIdx.x & 31;
  const int wave = threadIdx.x >> 5;
  const int wavesPerBlock = blockDim.x >> 5;
  const long long tile = (long long)blockIdx.x * wavesPerBlock + wave;
  const long long edgeBase = tile * 16;
  if (edgeBase >= E) return;           // uniform per wave: EXEC stays all-1s

  const int m = lane & 15;             // A row / B col owned by this lane
  const int kHalf = lane >> 4;         // which K-pair of the chunk

  long long e = edgeBase + m;
  long long ec = (e < E) ? e : (long long)(E - 1);  // clamp gathers for tail
  const v2f* kp = (const v2f*)ent + (size_t)dst[ec] * (D_DIM / 2);  // A rows (k)
  const v2f* qp = (const v2f*)ent + (size_t)src[ec] * (D_DIM / 2);  // B cols (qv)

  v8f c = {};
#pragma unroll
  for (int kk = 0; kk < D_DIM / 4; ++kk) {
    v2f a = kp[kk * 2 + kHalf];
    v2f b = qp[kk * 2 + kHalf];
    // 8 args: (neg_a, A, neg_b, B, c_mod, C, reuse_a, reuse_b)
    c = __builtin_amdgcn_wmma_f32_16x16x4_f32(
        false, a, false, b, (short)0, c, false, false);
  }

  // Extract diagonal: which (edge, component) does this lane own?
  int owner_e = -1, comp = 0;
  if (lane < 8)        { owner_e = lane;      comp = lane; }
  else if (lane >= 24) { owner_e = lane - 16; comp = lane - 24; }

  if (owner_e >= 0) {
    long long eo = edgeBase + owner_e;
    if (eo < E) {
      float v = c[0];
#pragma unroll
      for (int i = 1; i < 8; ++i) v = (comp == i) ? c[i] : v;
      float a = v * 0.0625f;           // 1/sqrt(128 + 128)
      alpha[eo] = a;
      atomicMax(&seg_max_ord[dst[eo]], f32_ord(a));
    }
  }
}

// ---------------------------------------------------------------------------
// K2: ealpha = exp(alpha - seg_max[dst]); seg_sum[dst] += ealpha (in place).
// ---------------------------------------------------------------------------
__global__ void exp_sum_kernel(const int* __restrict__ dst,
                               float* __restrict__ alpha,   // in: alpha, out: ealpha
                               const unsigned int* __restrict__ seg_max_ord,
                               float* __restrict__ seg_sum,
                               int E) {
  long long e = (long long)blockIdx.x * blockDim.x + threadIdx.x;
  if (e >= E) return;
  int d = dst[e];
  float mx = f32_unord(seg_max_ord[d]);
  float ea = __expf(alpha[e] - mx);
  alpha[e] = ea;
  atomicAdd(&seg_sum[d], ea);
}

// ---------------------------------------------------------------------------
// K3: out[dst[e]] += qv[e] * (ealpha[e] / seg_sum[dst[e]]).
// One wave per edge; each lane covers 4 dims (float4 gather + 4 f32 atomics).
// ---------------------------------------------------------------------------
__global__ void __launch_bounds__(256)
scatter_out_kernel(const float* __restrict__ ent,
                   const int* __restrict__ src,
                   const int* __restrict__ dst,
                   const float* __restrict__ ealpha,
                   const float* __restrict__ seg_sum,
                   float* __restrict__ out,
                   int E) {
  const int lane = threadIdx.x & 31;
  long long wave = ((long long)blockIdx.x * blockDim.x + threadIdx.x) >> 5;
  if (wave >= E) return;
  int e = (int)wave;
  int d = dst[e];
  float attn = ealpha[e] / seg_sum[d];
  const float4* q = (const float4*)(ent + (size_t)src[e] * D_DIM);
  float4 v = q[lane];
  float* o = out + (size_t)d * D_DIM + lane * 4;
  atomicAdd(o + 0, v.x * attn);
  atomicAdd(o + 1, v.y * attn);
  atomicAdd(o + 2, v.z * attn);
  atomicAdd(o + 3, v.w * attn);
}

// ---------------------------------------------------------------------------
// Host launcher. Inputs: [0]=entities (N*128 f32), [1]=relations (unused),
// [2]=edge_index (3*E int, rows: src,rel,dst). Output: N*128 f32.
// Workspace: seg_sum[N] f32 | seg_max_ord[N] u32 | alpha[E] f32.
// ---------------------------------------------------------------------------
extern "C" void kernel_launch(void* const* d_in, const int* in_sizes, int n_in,
                              void* d_out, int out_size, void* d_ws, size_t ws_size,
                              hipStream_t stream) {
  const float* entities = (const float*)d_in[0];
  const int* edge = (const int*)d_in[2];
  const int N = in_sizes[0] / D_DIM;
  const int E = in_sizes[2] / 3;
  const int* src = edge;                    // row 0
  const int* dst = edge + 2 * (size_t)E;    // row 2

  float* out = (float*)d_out;
  float* seg_sum = (float*)d_ws;
  unsigned int* seg_max_ord = (unsigned int*)(seg_sum + N);
  float* alpha = (float*)(seg_max_ord + N);

  const int BLK = 256;

  // K0: zero out + accumulators
  long long initTotal = (long long)N * D_DIM;
  int initBlocks = (int)((initTotal + BLK - 1) / BLK);
  init_ws_kernel<<<initBlocks, BLK, 0, stream>>>(out, seg_sum, seg_max_ord, N);

  // K1: WMMA alpha + segment max (16 edges per wave, 8 waves per block)
  long long tiles = ((long long)E + 15) / 16;
  int aBlocks = (int)((tiles + 7) / 8);
  alpha_wmma_kernel<<<aBlocks, BLK, 0, stream>>>(entities, src, dst, alpha,
                                                 seg_max_ord, E);

  // K2: exp + segment sum
  int eBlocks = (E + BLK - 1) / BLK;
  exp_sum_kernel<<<eBlocks, BLK, 0, stream>>>(dst, alpha, seg_max_ord, seg_sum, E);

  // K3: weighted scatter-add (1 wave per edge, 8 edges per block)
  int sBlocks = (E + 7) / 8;
  scatter_out_kernel<<<sBlocks, BLK, 0, stream>>>(entities, src, dst, alpha,
                                                  seg_sum, out, E);
}